// DeepMeshGcnLayer_68831145886184
// MI455X (gfx1250) — compile-verified
//
#include <hip/hip_runtime.h>

// Problem constants (match the reference)
#define NN 50000      // nodes
#define NE 600000     // edges
#define NG 64         // graphs
#define PD 128        // feature dim
#define HD 256        // MLP hidden dim
#define MSG_EPS 1e-7f
#define NORM_EPS 1e-5f

typedef __attribute__((ext_vector_type(2))) float v2f;
typedef __attribute__((ext_vector_type(8))) float v8f;

#if defined(__has_builtin)
#if __has_builtin(__builtin_amdgcn_wmma_f32_16x16x4_f32)
#define HAVE_WMMA_F32X4 1
#endif
#endif
#ifndef HAVE_WMMA_F32X4
#define HAVE_WMMA_F32X4 0
#endif

// One V_WMMA_F32_16X16X4_F32 step: D = A(16x4) * B(4x16) + C(16x16)
// A layout: lane&15 = M row; lane>>4 selects K-pair {0,1} or {2,3}; a.x/a.y = that pair.
// B layout: lane&15 = N col; same K-pair selection in b.x/b.y.
// C/D layout: VGPR i holds row (i + 8*(lane>>4)), col = lane&15.
__device__ __forceinline__ v8f wmma_f32x4(v2f a, v2f b, v8f c) {
#if HAVE_WMMA_F32X4
  return __builtin_amdgcn_wmma_f32_16x16x4_f32(false, a, false, b, (short)0, c,
                                               false, false);
#else
  // Fallback so the host pass / older toolchains still parse; not lane-exact.
  c[0] += a.x * b.x + a.y * b.y;
  return c;
#endif
}

__device__ __forceinline__ float leaky(float v) { return v > 0.f ? v : 0.01f * v; }

// Order-preserving float<->uint encoding for atomicMax on floats.
__device__ __forceinline__ unsigned fenc(float f) {
  unsigned u = __float_as_uint(f);
  return (u & 0x80000000u) ? ~u : (u | 0x80000000u);
}
__device__ __forceinline__ float fdec(unsigned u) {
  u = (u & 0x80000000u) ? (u & 0x7FFFFFFFu) : ~u;
  return __uint_as_float(u);
}

// ---------------- GraphNorm statistics ----------------
__global__ __launch_bounds__(256) void graph_sum_kernel(
    const float* __restrict__ x, const int* __restrict__ batch,
    float* __restrict__ gsum, float* __restrict__ gcnt) {
  int idx = blockIdx.x * blockDim.x + threadIdx.x;
  if (idx >= NN * PD) return;
  int n = idx >> 7, p = idx & 127;
  int b = batch[n];
  atomicAdd(&gsum[b * PD + p], x[idx]);
  if (p == 0) atomicAdd(&gcnt[b], 1.0f);
}

__global__ __launch_bounds__(256) void graph_var_kernel(
    const float* __restrict__ x, const int* __restrict__ batch,
    const float* __restrict__ ms, const float* __restrict__ gsum,
    const float* __restrict__ gcnt, float* __restrict__ gvar) {
  int idx = blockIdx.x * blockDim.x + threadIdx.x;
  if (idx >= NN * PD) return;
  int n = idx >> 7, p = idx & 127;
  int b = batch[n];
  float mean = gsum[b * PD + p] / gcnt[b];
  float sub = x[idx] - ms[p] * mean;
  atomicAdd(&gvar[b * PD + p], sub * sub);
}

__global__ __launch_bounds__(256) void compute_h_kernel(
    const float* __restrict__ x, const int* __restrict__ batch,
    const float* __restrict__ ms, const float* __restrict__ gw,
    const float* __restrict__ gb, const float* __restrict__ gsum,
    const float* __restrict__ gcnt, const float* __restrict__ gvar,
    float* __restrict__ h) {
  int idx = blockIdx.x * blockDim.x + threadIdx.x;
  if (idx >= NN * PD) return;
  int n = idx >> 7, p = idx & 127;
  int b = batch[n];
  float cnt = gcnt[b];
  float mean = gsum[b * PD + p] / cnt;
  float sub = x[idx] - ms[p] * mean;
  float rs = rsqrtf(gvar[b * PD + p] / cnt + NORM_EPS);
  h[idx] = leaky(gw[p] * sub * rs + gb[p]);
}

// ---------------- BatchNorm statistics over edges ----------------
__global__ __launch_bounds__(256) void bn_stats_kernel(
    const float* __restrict__ ea, float* __restrict__ esum,
    float* __restrict__ esumsq) {
  int p = threadIdx.x & 127;
  int half = threadIdx.x >> 7;  // 0 or 1
  int stride = gridDim.x * 2;
  float s = 0.f, s2 = 0.f;
  for (int e = blockIdx.x * 2 + half; e < NE; e += stride) {
    float v = ea[e * PD + p];
    s += v;
    s2 += v * v;
  }
  atomicAdd(&esum[p], s);
  atomicAdd(&esumsq[p], s2);
}

// ---------------- Edge linear: attr = leaky(bn(ea)) @ W^T + b ----------------
// 128 threads = 4 waves; block owns 64 edge rows; each wave one 16-row stripe.
// A (normalized edge rows) staged once in LDS; per column tile the 128x16 slab
// of pass_W is staged in LDS with coalesced loads, then WMMA reads both
// operands from LDS (A reads are loop-invariant but re-read from LDS for every column tile).
__global__ __launch_bounds__(128) void edge_gemm_kernel(
    const float* __restrict__ ea, const float* __restrict__ bn_gamma,
    const float* __restrict__ bn_beta, const float* __restrict__ esum,
    const float* __restrict__ esumsq, const float* __restrict__ passW,
    const float* __restrict__ passb, float* __restrict__ attr_out) {
  __shared__ float sA[64][PD + 4];
  __shared__ float sB[PD][17];  // stride 17: gcd(17,64)=1 -> conflict-free
  const int tid = threadIdx.x;
  const int e0 = blockIdx.x * 64;
  const float invE = 1.0f / (float)NE;
  for (int i = tid; i < 64 * PD; i += 128) {
    int r = i >> 7, c = i & 127;
    int e = e0 + r;
    float v = 0.f;
    if (e < NE) {
      float mu = esum[c] * invE;
      float var = esumsq[c] * invE - mu * mu;
      v = leaky(bn_gamma[c] * (ea[e * PD + c] - mu) * rsqrtf(var + NORM_EPS) +
                bn_beta[c]);
    }
    sA[r][c] = v;
  }
  const int lane = tid & 31;
  const int wave = tid >> 5;
  const int mn = lane & 15;            // M for A, N for B
  const int khalf = (lane >> 4) * 2;   // K-pair select
  const int arow = wave * 16 + mn;
  for (int ct = 0; ct < 8; ++ct) {
    __syncthreads();  // covers initial sA fill and sB reuse across iterations
    // Stage B[k][j] = W[ct*16+j][k]; consecutive threads read consecutive k.
    for (int i = tid; i < PD * 16; i += 128) {
      int j = i >> 7, k = i & 127;
      sB[k][j] = passW[(ct * 16 + j) * PD + k];
    }
    __syncthreads();
    v8f acc = {};
    for (int k = 0; k < PD; k += 4) {
      v2f a, b;
      a.x = sA[arow][k + khalf];
      a.y = sA[arow][k + khalf + 1];
      b.x = sB[k + khalf][mn];
      b.y = sB[k + khalf + 1][mn];
      acc = wmma_f32x4(a, b, acc);
    }
    const int cc = ct * 16 + mn;
    float bias = passb[cc];
#pragma unroll
    for (int i = 0; i < 8; ++i) {
      int r = wave * 16 + i + 8 * (lane >> 4);
      int e = e0 + r;
      if (e < NE) attr_out[e * PD + cc] = acc[i] + bias;
    }
  }
}

// ---------------- message passes (softmax over incoming edges) ----------------
__global__ __launch_bounds__(256) void msg_max_kernel(
    const int* __restrict__ ei, const float* __restrict__ h,
    const float* __restrict__ attr, const float* __restrict__ tptr,
    unsigned* __restrict__ menc) {
  int idx = blockIdx.x * blockDim.x + threadIdx.x;
  if (idx >= NE * PD) return;
  int e = idx >> 7, p = idx & 127;
  int s = ei[e], d = ei[NE + e];
  float m = fmaxf(h[s * PD + p] + attr[idx], 0.f) + MSG_EPS;
  float scaled = m * tptr[0];
  atomicMax(&menc[d * PD + p], fenc(scaled));
}

__global__ __launch_bounds__(256) void msg_sum_kernel(
    const int* __restrict__ ei, const float* __restrict__ h,
    const float* __restrict__ attr, const float* __restrict__ tptr,
    const unsigned* __restrict__ menc, float* __restrict__ denom,
    float* __restrict__ num) {
  int idx = blockIdx.x * blockDim.x + threadIdx.x;
  if (idx >= NE * PD) return;
  int e = idx >> 7, p = idx & 127;
  int s = ei[e], d = ei[NE + e];
  float m = fmaxf(h[s * PD + p] + attr[idx], 0.f) + MSG_EPS;
  float scaled = m * tptr[0];
  float w = __expf(scaled - fdec(menc[d * PD + p]));
  atomicAdd(&denom[d * PD + p], w);
  atomicAdd(&num[d * PD + p], m * w);
}

__global__ __launch_bounds__(256) void node_fin_kernel(
    const float* __restrict__ h, const float* __restrict__ denom,
    const float* __restrict__ num, float* __restrict__ xout) {
  int idx = blockIdx.x * blockDim.x + threadIdx.x;
  if (idx >= NN * PD) return;
  float den = denom[idx];
  float agg = den > 0.f ? num[idx] / den : 0.f;  // empty segments -> 0
  xout[idx] = agg + h[idx];
}

// ---------------- fused MLP: GEMM1 -> LayerNorm -> ReLU -> GEMM2 -> +x --------
// 256 threads = 8 waves; block owns 16 node rows kept in LDS. Waves cover
// disjoint column tiles so weights are read from L2 (W1/W2 are L2-resident).
__global__ __launch_bounds__(256) void mlp_kernel(
    const float* __restrict__ xout, const float* __restrict__ x,
    const float* __restrict__ W1, const float* __restrict__ lnG,
    const float* __restrict__ lnB, const float* __restrict__ W2,
    float* __restrict__ out1) {
  __shared__ float sX[16][PD + 4];
  __shared__ float sY[16][HD + 4];
  __shared__ float sMu[16], sRs[16];
  const int tid = threadIdx.x;
  const int n0 = blockIdx.x * 16;
  for (int i = tid; i < 16 * PD; i += 256) {
    int r = i >> 7, c = i & 127;
    int n = n0 + r;
    sX[r][c] = (n < NN) ? xout[n * PD + c] : 0.f;
  }
  __syncthreads();
  const int lane = tid & 31;
  const int wave = tid >> 5;
  const int mn = lane & 15;
  const int khalf = (lane >> 4) * 2;
  // y1 = sX @ W1^T : 16 x 256, each wave two 16-col tiles
#pragma unroll
  for (int t2 = 0; t2 < 2; ++t2) {
    int cc = (wave * 2 + t2) * 16 + mn;
    v8f acc = {};
    for (int k = 0; k < PD; k += 4) {
      v2f a, b;
      a.x = sX[mn][k + khalf];
      a.y = sX[mn][k + khalf + 1];
      b.x = W1[cc * PD + k + khalf];
      b.y = W1[cc * PD + k + khalf + 1];
      acc = wmma_f32x4(a, b, acc);
    }
#pragma unroll
    for (int i = 0; i < 8; ++i) sY[i + 8 * (lane >> 4)][cc] = acc[i];
  }
  __syncthreads();
  if (tid < 16) {
    float s = 0.f, s2 = 0.f;
    for (int c = 0; c < HD; ++c) {
      float v = sY[tid][c];
      s += v;
      s2 += v * v;
    }
    float mu = s * (1.0f / HD);
    float var = s2 * (1.0f / HD) - mu * mu;
    sMu[tid] = mu;
    sRs[tid] = rsqrtf(var + NORM_EPS);
  }
  __syncthreads();
  for (int i = tid; i < 16 * HD; i += 256) {
    int r = i >> 8, c = i & 255;
    float v = lnG[c] * (sY[r][c] - sMu[r]) * sRs[r] + lnB[c];
    sY[r][c] = fmaxf(v, 0.f);
  }
  __syncthreads();
  // y2 = relu(ln(y1)) @ W2^T : 16 x 128, each wave one 16-col tile, K=256
  {
    int cc = wave * 16 + mn;
    v8f acc = {};
    for (int k = 0; k < HD; k += 4) {
      v2f a, b;
      a.x = sY[mn][k + khalf];
      a.y = sY[mn][k + khalf + 1];
      b.x = W2[cc * HD + k + khalf];
      b.y = W2[cc * HD + k + khalf + 1];
      acc = wmma_f32x4(a, b, acc);
    }
#pragma unroll
    for (int i = 0; i < 8; ++i) {
      int r = i + 8 * (lane >> 4);
      int n = n0 + r;
      if (n < NN) out1[n * PD + cc] = acc[i] + x[n * PD + cc];
    }
  }
}

__global__ __launch_bounds__(256) void out2_add_kernel(
    const float* __restrict__ ea, float* __restrict__ out2) {
  int idx = blockIdx.x * blockDim.x + threadIdx.x;
  if (idx >= NE * PD) return;
  out2[idx] += ea[idx];
}

extern "C" void kernel_launch(void* const* d_in, const int* in_sizes, int n_in,
                              void* d_out, int out_size, void* d_ws,
                              size_t ws_size, hipStream_t stream) {
  (void)in_sizes; (void)n_in; (void)out_size; (void)ws_size;
  const float* x        = (const float*)d_in[0];
  const int*   ei       = (const int*)d_in[1];
  const float* ea       = (const float*)d_in[2];
  const int*   batch    = (const int*)d_in[3];
  const float* gw       = (const float*)d_in[4];
  const float* gb       = (const float*)d_in[5];
  const float* gms      = (const float*)d_in[6];
  const float* bn_gamma = (const float*)d_in[7];
  const float* bn_beta  = (const float*)d_in[8];
  const float* passW    = (const float*)d_in[9];
  const float* passb    = (const float*)d_in[10];
  const float* t        = (const float*)d_in[11];
  const float* W1       = (const float*)d_in[12];
  const float* lnG      = (const float*)d_in[13];
  const float* lnB      = (const float*)d_in[14];
  const float* W2       = (const float*)d_in[15];

  float* out1 = (float*)d_out;                       // [N,P]
  float* out2 = out1 + (size_t)NN * PD;              // [E,P]

  // Workspace layout
  float* h      = (float*)d_ws;                      // [N,P]
  float* xout   = h + (size_t)NN * PD;               // [N,P]
  float* accum  = xout + (size_t)NN * PD;            // zeroed region starts here
  unsigned* menc = (unsigned*)accum;                 // [N,P]
  float* denom  = accum + (size_t)NN * PD;           // [N,P]
  float* num    = denom + (size_t)NN * PD;           // [N,P]
  float* gsum   = num + (size_t)NN * PD;             // [G,P]
  float* gvar   = gsum + (size_t)NG * PD;            // [G,P]
  float* gcnt   = gvar + (size_t)NG * PD;            // [G]
  float* esum   = gcnt + NG;                         // [P]
  float* esumsq = esum + PD;                         // [P]
  size_t accum_bytes =
      ((size_t)3 * NN * PD + 2 * (size_t)NG * PD + NG + 2 * PD) * sizeof(float);
  hipMemsetAsync(accum, 0, accum_bytes, stream);

  const int npBlocks = (NN * PD + 255) / 256;
  const int epBlocks = (NE * PD + 255) / 256;

  graph_sum_kernel<<<npBlocks, 256, 0, stream>>>(x, batch, gsum, gcnt);
  graph_var_kernel<<<npBlocks, 256, 0, stream>>>(x, batch, gms, gsum, gcnt, gvar);
  compute_h_kernel<<<npBlocks, 256, 0, stream>>>(x, batch, gms, gw, gb, gsum,
                                                 gcnt, gvar, h);
  bn_stats_kernel<<<2048, 256, 0, stream>>>(ea, esum, esumsq);
  edge_gemm_kernel<<<(NE + 63) / 64, 128, 0, stream>>>(
      ea, bn_gamma, bn_beta, esum, esumsq, passW, passb, out2);
  msg_max_kernel<<<epBlocks, 256, 0, stream>>>(ei, h, out2, t, menc);
  msg_sum_kernel<<<epBlocks, 256, 0, stream>>>(ei, h, out2, t, menc, denom, num);
  node_fin_kernel<<<npBlocks, 256, 0, stream>>>(h, denom, num, xout);
  mlp_kernel<<<(NN + 15) / 16, 256, 0, stream>>>(xout, x, W1, lnG, lnB, W2, out1);
  out2_add_kernel<<<epBlocks, 256, 0, stream>>>(ea, out2);
}